// GNN_9749575762658
// MI455X (gfx1250) — compile-verified
//
#include <hip/hip_runtime.h>
#include <math.h>

#define N_NODES 100000
#define N_EDGES 2500000

typedef __attribute__((ext_vector_type(2))) float v2f;
typedef __attribute__((ext_vector_type(8))) float v8f;

// ---------------- degree / dinv ----------------

__global__ void k_zero_u32(unsigned* __restrict__ p, int n) {
  int i = blockIdx.x * blockDim.x + threadIdx.x;
  if (i < n) p[i] = 0u;
}

__global__ void k_deg_count(const int* __restrict__ dst, unsigned* __restrict__ cnt, int E) {
  int i = blockIdx.x * blockDim.x + threadIdx.x;
  if (i < E) atomicAdd(&cnt[dst[i]], 1u);  // native no-return global_atomic_add_u32
}

__global__ void k_dinv(float* __restrict__ dinv, int n) {
  int i = blockIdx.x * blockDim.x + threadIdx.x;
  if (i < n) {
    unsigned c = ((const unsigned*)dinv)[i];   // in-place: counts -> 1/sqrt(deg+1)
    dinv[i] = 1.0f / sqrtf((float)(c + 1u));
  }
}

// ---------------- dense GEMM via V_WMMA_F32_16X16X4_F32 ----------------
// One wave computes one 16x16 tile of H = A[Mx K] * W[K x N]. 8 waves/block.
// Fragment layouts (ISA 7.12.2, wave32):
//   A 16x4 f32 : lanes 0-15 M=lane   {v0=K k0,   v1=K k0+1}
//                lanes16-31 M=lane-16{v0=K k0+2, v1=K k0+3}
//   B  4x16 f32: lanes 0-15 N=lane   rows k0/k0+1 ; lanes 16-31 rows k0+2/k0+3
//   C/D 16x16  : VGPR r -> lanes 0-15 M=r, lanes 16-31 M=r+8
// K,N are compile-time: in-range steps have no selects at all; the K=10 tail
// uses clamped addresses + v_cndmask value-selects (no EXEC divergence).
template <int K, int N>
__global__ __launch_bounds__(256) void k_wmma_gemm(const float* __restrict__ A,
                                                   const float* __restrict__ W,
                                                   float* __restrict__ H,
                                                   int ntiles) {
  const int wave = threadIdx.x >> 5;
  const int tile = blockIdx.x * 8 + wave;
  if (tile >= ntiles) return;              // wave-uniform: EXEC stays all-ones
  constexpr int NTN = N >> 4;
  const int m0 = (tile / NTN) << 4;
  const int n0 = (tile % NTN) << 4;
  const int lane = threadIdx.x & 31;
  const int half = lane >> 4;              // 0 or 1
  const int idx  = lane & 15;              // M index for A, N index for B

  const float* __restrict__ arow = A + (m0 + idx) * K;
  const float* __restrict__ wcol = W + n0 + idx;

  v8f c = {};
  constexpr int KP = (K + 3) & ~3;
#pragma unroll
  for (int k0 = 0; k0 < KP; k0 += 4) {
    const int ka = k0 + half * 2;
    const bool v0 = (ka < K);
    const bool v1 = (ka + 1 < K);
    const int ka0 = v0 ? ka : 0;           // clamp -> unconditional loads
    const int ka1 = v1 ? ka + 1 : 0;
    float ax = arow[ka0];
    float ay = arow[ka1];
    float bx = wcol[ka0 * N];
    float by = wcol[ka1 * N];
    v2f a, b;
    a.x = v0 ? ax : 0.0f;  a.y = v1 ? ay : 0.0f;
    b.x = v0 ? bx : 0.0f;  b.y = v1 ? by : 0.0f;
    c = __builtin_amdgcn_wmma_f32_16x16x4_f32(false, a, false, b,
                                              (short)0, c, false, false);
  }
#pragma unroll
  for (int r = 0; r < 8; ++r) {
    const int m = m0 + r + half * 8;
    H[m * N + n0 + idx] = c[r];
  }
}

// ---------------- self-loop + bias init: out[i,f] = h[i,f]*dinv[i]^2 + b[f] ----------------

template <int LOGF>
__global__ void k_self_bias(const float* __restrict__ h, const float* __restrict__ dinv,
                            const float* __restrict__ bias, float* __restrict__ out,
                            int total) {
  int i = blockIdx.x * blockDim.x + threadIdx.x;
  if (i < total) {
    const int node = i >> LOGF;
    const int f = i & ((1 << LOGF) - 1);
    const float di = dinv[node];
    out[i] = h[i] * di * di + bias[f];
  }
}

// ---------------- edge scatter: out[dst,f] += h[src,f]*dinv[src]*dinv[dst] ----------------
// One lane per (edge, feature): F consecutive lanes per edge -> atomics land on
// consecutive addresses (1-2 cachelines/edge), index loads broadcast from cache.

__device__ __forceinline__ void gatomic_add_f32(float* p, float v) {
  asm volatile("global_atomic_add_f32 %0, %1, off" : : "v"(p), "v"(v) : "memory");
}

template <int LOGF>
__global__ void k_scatter(const int* __restrict__ src, const int* __restrict__ dst,
                          const float* __restrict__ dinv, const float* __restrict__ h,
                          float* __restrict__ out, int E) {
  const int gid = blockIdx.x * blockDim.x + threadIdx.x;
  const int e = gid >> LOGF;
  if (e >= E) return;
  const int f = gid & ((1 << LOGF) - 1);
  const int s = src[e];
  const int d = dst[e];
  const float nrm = dinv[s] * dinv[d];
  const float val = h[(s << LOGF) + f] * nrm;
  gatomic_add_f32(out + (d << LOGF) + f, val);
}

// ---------------- relu in place ----------------

__global__ void k_relu(float* __restrict__ p, int n) {
  int i = blockIdx.x * blockDim.x + threadIdx.x;
  if (i < n) p[i] = fmaxf(p[i], 0.0f);
}

// ---------------- host launcher ----------------

extern "C" void kernel_launch(void* const* d_in, const int* in_sizes, int n_in,
                              void* d_out, int out_size, void* d_ws, size_t ws_size,
                              hipStream_t stream) {
  const float* x  = (const float*)d_in[0];
  const int*   ei = (const int*)d_in[1];        // [2, E] row-major
  const float* W1 = (const float*)d_in[2];      // [10, 32]
  const float* b1 = (const float*)d_in[3];      // [32]
  const float* W2 = (const float*)d_in[4];      // [32, 16]
  const float* b2 = (const float*)d_in[5];      // [16]
  float* out = (float*)d_out;                   // [100000, 16]

  const int* src = ei;
  const int* dst = ei + N_EDGES;

  // workspace (floats): dinv[100000] | Abuf = h buffer [100000*32] | Bbuf = agg buffer [100000*32]
  float* dinv = (float*)d_ws;
  float* Abuf = dinv + N_NODES;
  float* Bbuf = Abuf + (size_t)N_NODES * 32;

  const int TB = 256;

  // 1) degree -> dinv
  k_zero_u32<<<(N_NODES + TB - 1) / TB, TB, 0, stream>>>((unsigned*)dinv, N_NODES);
  k_deg_count<<<(N_EDGES + TB - 1) / TB, TB, 0, stream>>>(dst, (unsigned*)dinv, N_EDGES);
  k_dinv<<<(N_NODES + TB - 1) / TB, TB, 0, stream>>>(dinv, N_NODES);

  // 2) h1 = x @ W1   (M=100000, K=10, N=32) -> Abuf
  {
    const int tiles = (N_NODES / 16) * (32 / 16);  // 12500
    k_wmma_gemm<10, 32><<<(tiles + 7) / 8, 256, 0, stream>>>(x, W1, Abuf, tiles);
  }

  // 3) Bbuf = h1*dinv^2 + b1 ; Bbuf[dst] += h1[src]*norm ; relu
  k_self_bias<5><<<(N_NODES * 32 + TB - 1) / TB, TB, 0, stream>>>(Abuf, dinv, b1, Bbuf,
                                                                  N_NODES * 32);
  k_scatter<5><<<(N_EDGES * 32 + TB - 1) / TB, TB, 0, stream>>>(src, dst, dinv, Abuf,
                                                                Bbuf, N_EDGES);
  k_relu<<<(N_NODES * 32 + TB - 1) / TB, TB, 0, stream>>>(Bbuf, N_NODES * 32);

  // 4) h2 = Bbuf @ W2   (M=100000, K=32, N=16) -> Abuf (reuse)
  {
    const int tiles = (N_NODES / 16) * (16 / 16);  // 6250
    k_wmma_gemm<32, 16><<<(tiles + 7) / 8, 256, 0, stream>>>(Bbuf, W2, Abuf, tiles);
  }

  // 5) out = h2*dinv^2 + b2 ; out[dst] += h2[src]*norm
  k_self_bias<4><<<(N_NODES * 16 + TB - 1) / TB, TB, 0, stream>>>(Abuf, dinv, b2, out,
                                                                  N_NODES * 16);
  k_scatter<4><<<(N_EDGES * 16 + TB - 1) / TB, TB, 0, stream>>>(src, dst, dinv, Abuf,
                                                                out, N_EDGES);
}